// Model_35021163332116
// MI455X (gfx1250) — compile-verified
//
#include <hip/hip_runtime.h>
#include <hip/hip_bf16.h>
#include <math.h>

typedef float v2f __attribute__((ext_vector_type(2)));
typedef float v8f __attribute__((ext_vector_type(8)));

#define BB 8
#define NN 2000
#define TT 24
#define HH 64
#define EE 32000
#define RR (BB*NN*TT)          /* 384000 rows of H=64 */
#define BNR (BB*NN)            /* 16000 */
#define ATT_SCALE 0.125f       /* 1/sqrt(64) */

#define GF_BIAS   1
#define GF_GELU   2
#define GF_ACCUM  4
#define GF_BTRANS 8

__device__ __forceinline__ float gelu_f(float v) {
  return 0.5f * v * (1.0f + erff(v * 0.70710678118654752f));
}

// ---------------------------------------------------------------------------
// fp32 WMMA GEMM, fully specialized: C[M,N] = act(alpha*A@B + bias (+C))
// One wave computes a 16(M) x (16*NT)(N) strip with V_WMMA_F32_16X16X4_F32.
// NT and FLAGS are compile-time -> k-loop body = NT unconditional WMMAs,
// no divergent control flow anywhere near the WMMAs (EXEC stays all-ones;
// the wave-bound early-exit is a scalar branch via readfirstlane).
// M % 16 == 0 and K % 4 == 0 at every call site.
// ---------------------------------------------------------------------------
template<int NT, int FLAGS>
__global__ __launch_bounds__(256)
void gemm_wmma_t(const float* __restrict__ A, const float* __restrict__ Bm,
                 const float* __restrict__ bias, float* __restrict__ C,
                 int M, int K, int lda, int ldb, int ldc,
                 int snStrips, int n0base, float alpha)
{
  int w  = (int)((blockIdx.x * 256u + threadIdx.x) >> 5);
  int tw = (M >> 4) * snStrips;
  if (__builtin_amdgcn_readfirstlane(w) >= tw) return;   // scalar branch, EXEC untouched
  int sm = w / snStrips, s = w % snStrips;
  int m0 = sm << 4, n0 = n0base + (s << 6);
  int lane = threadIdx.x & 31;
  int half = lane >> 4, l = lane & 15;
  const float* Arow = A + (size_t)(m0 + l) * lda + 2 * half;

  int ct[NT];
#pragma unroll
  for (int t = 0; t < NT; t++) ct[t] = n0 + t * 16 + l;

  v8f acc[NT];
#pragma unroll
  for (int t = 0; t < NT; t++) {
    if (FLAGS & GF_ACCUM) {
#pragma unroll
      for (int j = 0; j < 8; j++) acc[t][j] = C[(size_t)(m0 + j + 8 * half) * ldc + ct[t]];
    } else {
#pragma unroll
      for (int j = 0; j < 8; j++) acc[t][j] = 0.0f;
    }
  }

  const float* Bp[NT];
#pragma unroll
  for (int t = 0; t < NT; t++)
    Bp[t] = (FLAGS & GF_BTRANS) ? (Bm + (size_t)ct[t] * ldb + 2 * half)
                                : (Bm + (size_t)(2 * half) * ldb + ct[t]);

  for (int k = 0; k < K; k += 4) {
    v2f a; a.x = Arow[0]; a.y = Arow[1]; Arow += 4;
#pragma unroll
    for (int t = 0; t < NT; t++) {
      v2f b;
      if (FLAGS & GF_BTRANS) { b.x = Bp[t][0]; b.y = Bp[t][1];   Bp[t] += 4; }
      else                   { b.x = Bp[t][0]; b.y = Bp[t][ldb]; Bp[t] += 4 * (size_t)ldb; }
      acc[t] = __builtin_amdgcn_wmma_f32_16x16x4_f32(
          false, a, false, b, (short)0, acc[t], false, false);
    }
  }

#pragma unroll
  for (int t = 0; t < NT; t++) {
    float bs = (FLAGS & GF_BIAS) ? bias[ct[t]] : 0.0f;
#pragma unroll
    for (int j = 0; j < 8; j++) {
      float vv = acc[t][j] * alpha + bs;
      if (FLAGS & GF_GELU) vv = gelu_f(vv);
      C[(size_t)(m0 + j + 8 * half) * ldc + ct[t]] = vv;
    }
  }
}

// ---------------------------------------------------------------------------
// Fused masked-softmax(A) @ V per batch, WMMA. A[i,j] computed in-register:
//   a = smask[b,j] ? 0 : exp(scores[i,j] - mmax[b,i]);  out *= minv[b,i]
// ---------------------------------------------------------------------------
__global__ __launch_bounds__(256)
void attn_av_wmma(const float* __restrict__ S, const unsigned char* __restrict__ smask,
                  const float* __restrict__ mmax, const float* __restrict__ minv,
                  const float* __restrict__ V, float* __restrict__ O)
{
  int w = (int)((blockIdx.x * 256u + threadIdx.x) >> 5);
  const int wpb = NN / 16;                   // 125
  if (__builtin_amdgcn_readfirstlane(w) >= BB * wpb) return;
  int b = w / wpb, sm = w % wpb;
  int m0 = sm * 16;
  int lane = threadIdx.x & 31;
  int half = lane >> 4, l = lane & 15;
  int r = m0 + l;
  const float* Srow = S + (size_t)r * NN + 2 * half;
  const unsigned char* msk = smask + (size_t)b * NN + 2 * half;
  float mx = mmax[(size_t)b * NN + r];
  const float* Vb = V + (size_t)b * NN * HH + (size_t)(2 * half) * HH;

  v8f acc[4];
#pragma unroll
  for (int t = 0; t < 4; t++)
#pragma unroll
    for (int j = 0; j < 8; j++) acc[t][j] = 0.0f;

  for (int k = 0; k < NN; k += 4) {
    v2f a;
    a.x = msk[0] ? 0.0f : __expf(Srow[0] - mx);
    a.y = msk[1] ? 0.0f : __expf(Srow[1] - mx);
    Srow += 4; msk += 4;
#pragma unroll
    for (int t = 0; t < 4; t++) {
      int ctl = t * 16 + l;
      v2f bv;
      bv.x = Vb[ctl];
      bv.y = Vb[HH + ctl];
      acc[t] = __builtin_amdgcn_wmma_f32_16x16x4_f32(
          false, a, false, bv, (short)0, acc[t], false, false);
    }
    Vb += 4 * HH;
  }
#pragma unroll
  for (int t = 0; t < 4; t++) {
    int ctl = t * 16 + l;
#pragma unroll
    for (int j = 0; j < 8; j++) {
      int i = m0 + j + 8 * half;
      O[((size_t)b * NN + i) * HH + ctl] = acc[t][j] * minv[(size_t)b * NN + i];
    }
  }
}

// ---------------------------------------------------------------------------
// Naive GEMM for tiny K / tiny N (in_lin K=3, t_time K=5, stage2 N=1)
// ---------------------------------------------------------------------------
__global__ void gemm_naive(const float* __restrict__ A, const float* __restrict__ Bm,
                           const float* __restrict__ bias, float* __restrict__ C,
                           int M, int N, int K, int flags)
{
  size_t idx = (size_t)blockIdx.x * blockDim.x + threadIdx.x;
  if (idx >= (size_t)M * N) return;
  int c = (int)(idx % N); size_t r = idx / N;
  float acc = 0.0f;
  for (int k = 0; k < K; k++) acc += A[r * K + k] * Bm[(size_t)k * N + c];
  if (flags & GF_BIAS) acc += bias[c];
  if (flags & GF_GELU) acc = gelu_f(acc);
  C[idx] = acc;
}

// ------------------------- elementwise / small kernels ----------------------
__global__ void add_pe(const float* __restrict__ XF, float* __restrict__ O)
{
  size_t idx = (size_t)blockIdx.x * blockDim.x + threadIdx.x;
  if (idx >= (size_t)RR * HH) return;
  int c = (int)(idx & 63); size_t r = idx >> 6;
  int t = (int)(r % TT);
  int ci = c & ~1;
  float div = __expf(-logf(10000.0f) * (float)ci / (float)HH);
  float ang = (float)t * div;
  float pe = (c & 1) ? cosf(ang) : sinf(ang);
  O[idx] = XF[idx] + pe;
}

__global__ void combine_tx(float* __restrict__ TX, const float* __restrict__ TEP,
                           const float* __restrict__ NEP, const float* __restrict__ tb)
{
  size_t idx = (size_t)blockIdx.x * blockDim.x + threadIdx.x;
  if (idx >= (size_t)RR * HH) return;
  int c = (int)(idx & 63); size_t r = idx >> 6;
  int t = (int)(r % TT);
  size_t bn = r / TT;
  int n = (int)(bn % NN);
  int b = (int)(bn / NN);
  float v = TX[idx] + TEP[((size_t)b * TT + t) * HH + c] + NEP[(size_t)n * HH + c] + tb[c];
  TX[idx] = gelu_f(v);
}

__global__ void fuse_gelu_add(const float* __restrict__ Xa, const float* __restrict__ Xb,
                              float* __restrict__ O)
{
  size_t idx = (size_t)blockIdx.x * blockDim.x + threadIdx.x;
  if (idx >= (size_t)RR * HH) return;
  O[idx] = gelu_f(Xa[idx] + Xb[idx]);
}

__global__ void mask_select(float* __restrict__ O, const float* __restrict__ XF,
                            const unsigned char* __restrict__ mask)
{
  size_t idx = (size_t)blockIdx.x * blockDim.x + threadIdx.x;
  if (idx >= (size_t)RR * HH) return;
  size_t r = idx >> 6;
  if (!mask[r]) O[idx] = XF[idx];
}

__global__ void gelu_mask_select(const float* __restrict__ FIN, const float* __restrict__ XF,
                                 const unsigned char* __restrict__ mask, float* __restrict__ O)
{
  size_t idx = (size_t)blockIdx.x * blockDim.x + threadIdx.x;
  if (idx >= (size_t)RR * HH) return;
  size_t r = idx >> 6;
  O[idx] = mask[r] ? gelu_f(FIN[idx]) : XF[idx];
}

__global__ void mul_conf(float* __restrict__ Y, const float* __restrict__ CONF)
{
  size_t idx = (size_t)blockIdx.x * blockDim.x + threadIdx.x;
  if (idx >= (size_t)RR * HH) return;
  Y[idx] *= CONF[idx >> 6];
}

__global__ void conf_kernel(const unsigned char* __restrict__ mask,
                            const float* __restrict__ SCONF, const float* __restrict__ TCONF,
                            float* __restrict__ CONF)
{
  size_t idx = (size_t)blockIdx.x * blockDim.x + threadIdx.x;
  if (idx >= (size_t)RR) return;
  CONF[idx] = mask[idx] ? 0.5f * (SCONF[idx / TT] + TCONF[idx]) : 1.0f;
}

__global__ void band_kernel(const float* __restrict__ Y, float* __restrict__ O)
{
  size_t idx = (size_t)blockIdx.x * blockDim.x + threadIdx.x;
  if (idx >= (size_t)RR * HH) return;
  size_t r = idx >> 6;
  int t = (int)(r % TT);
  float sum = 0.0f; int cnt = 0;
  if (t > 0)      { sum += Y[idx - HH]; cnt++; }
  if (t < TT - 1) { sum += Y[idx + HH]; cnt++; }
  O[idx] = sum / (float)cnt;
}

__global__ void gather64(const float* __restrict__ Emb, const int* __restrict__ ids,
                         float* __restrict__ O, int n)
{
  size_t idx = (size_t)blockIdx.x * blockDim.x + threadIdx.x;
  if (idx >= (size_t)n * HH) return;
  int h = (int)(idx & 63); int i = (int)(idx >> 6);
  O[idx] = Emb[(size_t)ids[i] * HH + h];
}

// --------------------------- CSR build (deterministic) ----------------------
__global__ void zero_i32(int* p, int n)
{
  int idx = blockIdx.x * blockDim.x + threadIdx.x;
  if (idx < n) p[idx] = 0;
}
__global__ void count_indeg(const int* __restrict__ ei, int* __restrict__ ideg)
{
  int e = blockIdx.x * blockDim.x + threadIdx.x;
  if (e >= EE) return;
  atomicAdd(&ideg[ei[EE + e]], 1);
}
__global__ void cnt_float(const int* __restrict__ ideg, float* __restrict__ cnt)
{
  int i = blockIdx.x * blockDim.x + threadIdx.x;
  if (i >= NN) return;
  int d = ideg[i];
  cnt[i] = (float)(d < 1 ? 1 : d);
}
__global__ void build_csr(const int* __restrict__ ei, const int* __restrict__ ideg,
                          int* __restrict__ rowptr, int* __restrict__ tmpoff,
                          int* __restrict__ col)
{
  if (blockIdx.x || threadIdx.x) return;     // single-thread: replay-deterministic
  int acc = 0;
  for (int i = 0; i < NN; i++) { rowptr[i] = acc; tmpoff[i] = acc; acc += ideg[i]; }
  rowptr[NN] = acc;
  const int* src = ei; const int* dst = ei + EE;
  for (int e = 0; e < EE; e++) col[tmpoff[dst[e]]++] = src[e];
}

__global__ void gather_mean_nodes(const float* __restrict__ NF, const int* __restrict__ rowptr,
                                  const int* __restrict__ col, const float* __restrict__ cnt,
                                  float* __restrict__ O)
{
  size_t idx = (size_t)blockIdx.x * blockDim.x + threadIdx.x;
  if (idx >= (size_t)NN * HH) return;
  int h = (int)(idx & 63); int i = (int)(idx >> 6);
  float s = 0.0f;
  int beg = rowptr[i], end = rowptr[i + 1];
  for (int e = beg; e < end; e++) s += NF[(size_t)col[e] * HH + h];
  O[idx] = s / cnt[i];
}

__global__ void gather_mean_bt(const float* __restrict__ X, const int* __restrict__ rowptr,
                               const int* __restrict__ col, const float* __restrict__ cnt,
                               float* __restrict__ O)
{
  size_t idx = (size_t)blockIdx.x * blockDim.x + threadIdx.x;
  if (idx >= (size_t)RR * HH) return;
  int h = (int)(idx & 63); size_t row = idx >> 6;
  int t = (int)(row % TT);
  size_t bn = row / TT;
  int n = (int)(bn % NN);
  int b = (int)(bn / NN);
  float s = 0.0f;
  int beg = rowptr[n], end = rowptr[n + 1];
  for (int e = beg; e < end; e++)
    s += X[(((size_t)b * NN + col[e]) * TT + t) * HH + h];
  O[idx] = s / cnt[n];
}

// --------------------------- softmax stat reductions ------------------------
__global__ __launch_bounds__(256)
void pstats_kernel(const float* __restrict__ S, float* __restrict__ pmax, float* __restrict__ psum)
{
  __shared__ float red[256];
  int i = blockIdx.x, tid = threadIdx.x;
  const float* row = S + (size_t)i * NN;
  float m = -INFINITY;
  for (int j = tid; j < NN; j += 256) m = fmaxf(m, row[j]);
  red[tid] = m; __syncthreads();
  for (int s = 128; s > 0; s >>= 1) { if (tid < s) red[tid] = fmaxf(red[tid], red[tid + s]); __syncthreads(); }
  float mx = red[0]; __syncthreads();
  float sum = 0.0f;
  for (int j = tid; j < NN; j += 256) sum += __expf(row[j] - mx);
  red[tid] = sum; __syncthreads();
  for (int s = 128; s > 0; s >>= 1) { if (tid < s) red[tid] += red[tid + s]; __syncthreads(); }
  if (tid == 0) { pmax[i] = mx; psum[i] = red[0]; }
}

__global__ __launch_bounds__(256)
void smask_kernel(const unsigned char* __restrict__ mask, unsigned char* __restrict__ smask)
{
  int idx = blockIdx.x * blockDim.x + threadIdx.x;
  if (idx >= BB * NN) return;
  const unsigned char* m = mask + (size_t)idx * TT;
  unsigned char a = 1;
  for (int t = 0; t < TT; t++) a = (unsigned char)(a && m[t]);
  smask[idx] = a;
}

__global__ __launch_bounds__(256)
void sconf_kernel(const float* __restrict__ S, const unsigned char* __restrict__ smask,
                  const float* __restrict__ pmax, const float* __restrict__ psum,
                  float* __restrict__ sconf)
{
  __shared__ float red[256];
  int bi = blockIdx.x, tid = threadIdx.x;
  int b = bi / NN, i = bi % NN;
  const float* row = S + (size_t)i * NN;
  const unsigned char* msk = smask + (size_t)b * NN;
  float mx = pmax[i];
  float sum = 0.0f;
  for (int j = tid; j < NN; j += 256) if (msk[j]) sum += __expf(row[j] - mx);
  red[tid] = sum; __syncthreads();
  for (int s = 128; s > 0; s >>= 1) { if (tid < s) red[tid] += red[tid + s]; __syncthreads(); }
  if (tid == 0) sconf[bi] = 1.0f - red[0] / psum[i];
}

__global__ __launch_bounds__(256)
void mstats_kernel(const float* __restrict__ S, const unsigned char* __restrict__ smask,
                   float* __restrict__ mmax, float* __restrict__ minv)
{
  __shared__ float red[256];
  int bi = blockIdx.x, tid = threadIdx.x;
  int b = bi / NN, i = bi % NN;
  const float* row = S + (size_t)i * NN;
  const unsigned char* msk = smask + (size_t)b * NN;
  float m = -INFINITY;
  for (int j = tid; j < NN; j += 256) if (!msk[j]) m = fmaxf(m, row[j]);
  red[tid] = m; __syncthreads();
  for (int s = 128; s > 0; s >>= 1) { if (tid < s) red[tid] = fmaxf(red[tid], red[tid + s]); __syncthreads(); }
  float mx = red[0]; __syncthreads();
  float sum = 0.0f;
  if (mx != -INFINITY)
    for (int j = tid; j < NN; j += 256) if (!msk[j]) sum += __expf(row[j] - mx);
  red[tid] = sum; __syncthreads();
  for (int s = 128; s > 0; s >>= 1) { if (tid < s) red[tid] += red[tid + s]; __syncthreads(); }
  if (tid == 0) {
    mmax[bi] = (mx == -INFINITY) ? 0.0f : mx;
    minv[bi] = (red[0] > 0.0f) ? 1.0f / red[0] : 0.0f;   // nan_to_num -> 0 rows
  }
}

// ------------------------------ temporal attention ---------------------------
__global__ __launch_bounds__(128)
void temporal_attn(const float* __restrict__ TQp, const float* __restrict__ TKp,
                   const float* __restrict__ TVp, const unsigned char* __restrict__ mask,
                   float* __restrict__ TATTp, float* __restrict__ TCONFp)
{
  __shared__ float sQ[TT * HH], sK[TT * HH], sV[TT * HH], sA[TT * TT];
  int bn = blockIdx.x, tid = threadIdx.x;
  size_t base = (size_t)bn * TT * HH;
  for (int i = tid; i < TT * HH; i += 128) { sQ[i] = TQp[base + i]; sK[i] = TKp[base + i]; sV[i] = TVp[base + i]; }
  __syncthreads();
  for (int i = tid; i < TT * TT; i += 128) {
    int s = i / TT, t = i % TT;
    float acc = 0.0f;
    for (int f = 0; f < HH; f++) acc += sQ[s * HH + f] * sK[t * HH + f];
    sA[i] = acc * ATT_SCALE;
  }
  __syncthreads();
  if (tid < TT) {
    int s = tid;
    float mx = -INFINITY;
    for (int t = 0; t < TT; t++) mx = fmaxf(mx, sA[s * TT + t]);
    float sum = 0.0f;
    for (int t = 0; t < TT; t++) { float e = __expf(sA[s * TT + t] - mx); sA[s * TT + t] = e; sum += e; }
    float inv = 1.0f / sum;
    float conf = 1.0f;
    const unsigned char* mrow = mask + (size_t)bn * TT;
    for (int t = 0; t < TT; t++) { float p = sA[s * TT + t] * inv; sA[s * TT + t] = p; if (mrow[t]) conf -= p; }
    TCONFp[(size_t)bn * TT + s] = conf;
  }
  __syncthreads();
  for (int i = tid; i < TT * HH; i += 128) {
    int s = i / HH, f = i % HH;
    float acc = 0.0f;
    for (int t = 0; t < TT; t++) acc += sA[s * TT + t] * sV[t * HH + f];
    TATTp[base + i] = acc;
  }
}

// ------------------------------ host-side dispatch ---------------------------
template<int NT>
static void launch_nt(hipStream_t st, int flags, int blocks,
                      const float* A, const float* B, const float* bias, float* C,
                      int M, int K, int lda, int ldb, int ldc,
                      int sn, int n0, float alpha)
{
  switch (flags) {
    case 0:
      gemm_wmma_t<NT, 0><<<blocks, 256, 0, st>>>(A, B, bias, C, M, K, lda, ldb, ldc, sn, n0, alpha); break;
    case GF_BIAS:
      gemm_wmma_t<NT, GF_BIAS><<<blocks, 256, 0, st>>>(A, B, bias, C, M, K, lda, ldb, ldc, sn, n0, alpha); break;
    case GF_BIAS | GF_GELU:
      gemm_wmma_t<NT, GF_BIAS | GF_GELU><<<blocks, 256, 0, st>>>(A, B, bias, C, M, K, lda, ldb, ldc, sn, n0, alpha); break;
    case GF_ACCUM:
      gemm_wmma_t<NT, GF_ACCUM><<<blocks, 256, 0, st>>>(A, B, bias, C, M, K, lda, ldb, ldc, sn, n0, alpha); break;
    case GF_ACCUM | GF_BIAS | GF_GELU:
      gemm_wmma_t<NT, GF_ACCUM | GF_BIAS | GF_GELU><<<blocks, 256, 0, st>>>(A, B, bias, C, M, K, lda, ldb, ldc, sn, n0, alpha); break;
    case GF_BTRANS:
      gemm_wmma_t<NT, GF_BTRANS><<<blocks, 256, 0, st>>>(A, B, bias, C, M, K, lda, ldb, ldc, sn, n0, alpha); break;
    default:
      gemm_wmma_t<NT, GF_BIAS><<<blocks, 256, 0, st>>>(A, B, bias, C, M, K, lda, ldb, ldc, sn, n0, alpha); break;
  }
}

static void gemm(hipStream_t st, const float* A, const float* B, const float* bias, float* C,
                 int M, int N, int K, int lda, int ldb, int ldc, int flags, float alpha)
{
  int snFull = N >> 6;
  int remT   = (N & 63) >> 4;                 // remaining 16-wide tiles (0..3)
  if (snFull) {
    int waves = (M >> 4) * snFull;
    launch_nt<4>(st, flags, (waves + 7) >> 3, A, B, bias, C, M, K, lda, ldb, ldc, snFull, 0, alpha);
  }
  if (remT) {
    int waves = M >> 4;
    int blocks = (waves + 7) >> 3;
    int n0 = snFull << 6;
    if (remT == 1)      launch_nt<1>(st, flags, blocks, A, B, bias, C, M, K, lda, ldb, ldc, 1, n0, alpha);
    else if (remT == 2) launch_nt<2>(st, flags, blocks, A, B, bias, C, M, K, lda, ldb, ldc, 1, n0, alpha);
    else                launch_nt<3>(st, flags, blocks, A, B, bias, C, M, K, lda, ldb, ldc, 1, n0, alpha);
  }
}

#define NBLK(n) ((unsigned)(((size_t)(n) + 255) / 256))

extern "C" void kernel_launch(void* const* d_in, const int* in_sizes, int n_in,
                              void* d_out, int out_size, void* d_ws, size_t ws_size,
                              hipStream_t stream)
{
  // ---- inputs ----
  const int*            node_embed = (const int*)d_in[0];
  const float*          x          = (const float*)d_in[1];
  const float*          ex         = (const float*)d_in[2];
  const int*            edge_index = (const int*)d_in[3];
  const unsigned char*  mask       = (const unsigned char*)d_in[4];
  const float* P[41];
  for (int i = 0; i < 41; i++) P[i] = (const float*)d_in[5 + i];
  const float *in_lin_W=P[0], *in_lin_b=P[1], *s_in_W=P[2], *s_in_b=P[3], *s_node_emb=P[4],
    *s_gnn_rel_W=P[5], *s_gnn_rel_b=P[6], *s_gnn_root_W=P[7],
    *s_q_W=P[8], *s_q_b=P[9], *s_k_W=P[10], *s_k_b=P[11], *s_v_W=P[12], *s_v_b=P[13],
    *s_out_W=P[14], *s_out_b=P[15], *t_in_W=P[16], *t_in_b=P[17],
    *t_time_W=P[18], *t_time_b=P[19], *t_node_emb=P[20],
    *t_q_W=P[21], *t_q_b=P[22], *t_k_W=P[23], *t_k_b=P[24], *t_v_W=P[25], *t_v_b=P[26],
    *t_out_W=P[27], *t_out_b=P[28], *stage1_W=P[29], *stage1_b=P[30],
    *stage2_W=P[31], *stage2_b=P[32], *p_lin_W=P[33], *p_lin_b=P[34],
    *p_lins_W=P[35], *p_lins_b=P[36], *p_lint_W=P[37], *p_lint_b=P[38],
    *p_out_W=P[39], *p_out_b=P[40];

  // ---- workspace carve ----
  char* base = (char*)d_ws;
  auto alloc = [&](size_t nbytes) -> char* {
    char* p = base; base += (nbytes + 255) & ~(size_t)255; return p;
  };
  const size_t BIG = (size_t)RR * HH * sizeof(float);
  float* XF   = (float*)alloc(BIG);
  float* XFPE = (float*)alloc(BIG);   // later aliased: TATT
  float* TXP  = (float*)alloc(BIG);   // part1 -> tx -> TREC
  float* TQ   = (float*)alloc(BIG);   // later: G
  float* TK   = (float*)alloc(BIG);   // later: OUT
  float* TV   = (float*)alloc(BIG);   // later: OUT2
  float* SREC = (float*)alloc(BIG);
  float* PA   = (float*)alloc(BIG);
  float* PB   = (float*)alloc(BIG);
  float* PC   = (float*)alloc(BIG);
  float* PD   = (float*)alloc(BIG);   // FIN
  float* SX    = (float*)alloc((size_t)BNR * HH * 4);
  float* V     = (float*)alloc((size_t)BNR * HH * 4);
  float* SATT  = (float*)alloc((size_t)BNR * HH * 4);
  float* NFa   = (float*)alloc((size_t)NN * HH * 4);
  float* NFb   = (float*)alloc((size_t)NN * HH * 4);
  float* AGG   = (float*)alloc((size_t)NN * HH * 4);
  float* Qb    = (float*)alloc((size_t)NN * HH * 4);
  float* Kb    = (float*)alloc((size_t)NN * HH * 4);
  float* NEG   = (float*)alloc((size_t)NN * HH * 4);
  float* NEP   = (float*)alloc((size_t)NN * HH * 4);
  float* SCORES= (float*)alloc((size_t)NN * NN * 4);
  float* PMAX  = (float*)alloc((size_t)NN * 4);
  float* PSUM  = (float*)alloc((size_t)NN * 4);
  unsigned char* SMASK = (unsigned char*)alloc((size_t)BB * NN);
  float* SCONF = (float*)alloc((size_t)BB * NN * 4);
  float* MMAX  = (float*)alloc((size_t)BB * NN * 4);
  float* MINV  = (float*)alloc((size_t)BB * NN * 4);
  float* TELIN = (float*)alloc((size_t)BB * TT * HH * 4);
  float* TEP   = (float*)alloc((size_t)BB * TT * HH * 4);
  float* TCONF = (float*)alloc((size_t)RR * 4);
  float* CONF  = (float*)alloc((size_t)RR * 4);
  float* CNT   = (float*)alloc((size_t)NN * 4);
  int*   IDEG  = (int*)alloc((size_t)NN * 4);
  int*   ROWPTR= (int*)alloc((size_t)(NN + 1) * 4);
  int*   TMPOFF= (int*)alloc((size_t)(NN + 1) * 4);
  int*   COL   = (int*)alloc((size_t)EE * 4);
  (void)ws_size; (void)n_in; (void)in_sizes; (void)out_size;

  // 1. xf = gelu(x @ in_lin_W + b)          [R,3]@[3,64]
  gemm_naive<<<NBLK((size_t)RR * HH), 256, 0, stream>>>(x, in_lin_W, in_lin_b, XF, RR, HH, 3, GF_BIAS | GF_GELU);
  // 2. xfpe = xf + PE
  add_pe<<<NBLK((size_t)RR * HH), 256, 0, stream>>>(XF, XFPE);
  // 3. sx = gelu(xf.view(BN,1536) @ s_in_W + b)
  gemm(stream, XF, s_in_W, s_in_b, SX, BNR, HH, TT * HH, TT * HH, HH, HH, GF_BIAS | GF_GELU, 1.0f);

  // 4. CSR (deterministic) + cnt
  zero_i32<<<NBLK(NN), 256, 0, stream>>>(IDEG, NN);
  count_indeg<<<NBLK(EE), 256, 0, stream>>>(edge_index, IDEG);
  cnt_float<<<NBLK(NN), 256, 0, stream>>>(IDEG, CNT);
  build_csr<<<1, 1, 0, stream>>>(edge_index, IDEG, ROWPTR, TMPOFF, COL);

  // 5. nf = s_node_emb[node_embed]
  gather64<<<NBLK((size_t)NN * HH), 256, 0, stream>>>(s_node_emb, node_embed, NFa, NN);
  // 6. GNN x2 (shared weights): nf = gelu(agg@rel + rel_b + nf@root)
  float* nfp = NFa; float* nft = NFb;
  for (int l = 0; l < 2; l++) {
    gather_mean_nodes<<<NBLK((size_t)NN * HH), 256, 0, stream>>>(nfp, ROWPTR, COL, CNT, AGG);
    gemm(stream, nfp, s_gnn_root_W, nullptr, nft, NN, HH, HH, HH, HH, HH, 0, 1.0f);
    gemm(stream, AGG, s_gnn_rel_W, s_gnn_rel_b, nft, NN, HH, HH, HH, HH, HH,
         GF_ACCUM | GF_BIAS | GF_GELU, 1.0f);
    float* tmp = nfp; nfp = nft; nft = tmp;
  }
  // 7. q, k, v
  gemm(stream, nfp, s_q_W, s_q_b, Qb, NN, HH, HH, HH, HH, HH, GF_BIAS, 1.0f);
  gemm(stream, nfp, s_k_W, s_k_b, Kb, NN, HH, HH, HH, HH, HH, GF_BIAS, 1.0f);
  gemm(stream, SX, s_v_W, s_v_b, V, BNR, HH, HH, HH, HH, HH, GF_BIAS, 1.0f);
  // 8. scores = (q @ k^T) * SCALE   (N=2000 -> 31 full strips + one 16-wide tile)
  gemm(stream, Qb, Kb, nullptr, SCORES, NN, NN, HH, HH, HH, NN, GF_BTRANS, ATT_SCALE);
  // 9. masks + softmax stats + s_conf
  smask_kernel<<<NBLK(BB * NN), 256, 0, stream>>>(mask, SMASK);
  pstats_kernel<<<NN, 256, 0, stream>>>(SCORES, PMAX, PSUM);
  sconf_kernel<<<BB * NN, 256, 0, stream>>>(SCORES, SMASK, PMAX, PSUM, SCONF);
  mstats_kernel<<<BB * NN, 256, 0, stream>>>(SCORES, SMASK, MMAX, MINV);
  // 10. s_att = softmax(masked scores) @ v   (fused, WMMA)
  attn_av_wmma<<<(BB * (NN / 16) + 7) / 8, 256, 0, stream>>>(SCORES, SMASK, MMAX, MINV, V, SATT);
  // 11. s_rec = s_att @ s_out_W + b   [BN,64]@[64,1536]
  gemm(stream, SATT, s_out_W, s_out_b, SREC, BNR, TT * HH, HH, HH, TT * HH, TT * HH, GF_BIAS, 1.0f);

  // 12-16. tx = gelu(concat([xfpe, te, ne]) @ t_in_W + b) via split GEMMs
  gemm_naive<<<NBLK((size_t)BB * TT * HH), 256, 0, stream>>>(ex, t_time_W, t_time_b, TELIN, BB * TT, HH, 5, GF_BIAS);
  gemm(stream, TELIN, t_in_W + 64 * 64, nullptr, TEP, BB * TT, HH, HH, HH, HH, HH, 0, 1.0f);
  gather64<<<NBLK((size_t)NN * HH), 256, 0, stream>>>(t_node_emb, node_embed, NEG, NN);
  gemm(stream, NEG, t_in_W + 128 * 64, nullptr, NEP, NN, HH, HH, HH, HH, HH, 0, 1.0f);
  gemm(stream, XFPE, t_in_W, nullptr, TXP, RR, HH, HH, HH, HH, HH, 0, 1.0f);
  combine_tx<<<NBLK((size_t)RR * HH), 256, 0, stream>>>(TXP, TEP, NEP, t_in_b);
  // 17. tq, tk, tv
  gemm(stream, TXP, t_q_W, t_q_b, TQ, RR, HH, HH, HH, HH, HH, GF_BIAS, 1.0f);
  gemm(stream, TXP, t_k_W, t_k_b, TK, RR, HH, HH, HH, HH, HH, GF_BIAS, 1.0f);
  gemm(stream, TXP, t_v_W, t_v_b, TV, RR, HH, HH, HH, HH, HH, GF_BIAS, 1.0f);
  // 18. temporal attention (TATT aliases XFPE)
  float* TATT = XFPE;
  temporal_attn<<<BB * NN, 128, 0, stream>>>(TQ, TK, TV, mask, TATT, TCONF);
  // 19. t_rec = t_att @ t_out_W + b  (into TXP)
  gemm(stream, TATT, t_out_W, t_out_b, TXP, RR, HH, HH, HH, HH, HH, GF_BIAS, 1.0f);

  // 20-23. fusion
  float* G = TQ;
  fuse_gelu_add<<<NBLK((size_t)RR * HH), 256, 0, stream>>>(SREC, TXP, G);
  float* OUT = TK; float* OUT2 = TV;
  gemm(stream, G, stage1_W, stage1_b, OUT, RR, HH, HH, HH, HH, HH, GF_BIAS, 1.0f);
  mask_select<<<NBLK((size_t)RR * HH), 256, 0, stream>>>(OUT, XF, mask);
  conf_kernel<<<NBLK(RR), 256, 0, stream>>>(mask, SCONF, TCONF, CONF);

  // 24. propagation x2 (shared weights)
  float* oin = OUT; float* oout = OUT2;
  for (int l = 0; l < 2; l++) {
    // temporal branch: t_xf = BAND @ (lint(out) * conf)
    gemm(stream, oin, p_lint_W, p_lint_b, PA, RR, HH, HH, HH, HH, HH, GF_BIAS, 1.0f);
    mul_conf<<<NBLK((size_t)RR * HH), 256, 0, stream>>>(PA, CONF);
    band_kernel<<<NBLK((size_t)RR * HH), 256, 0, stream>>>(PA, PB);
    // spatial branch: s_xf = scatter-mean(lins(out) * conf)
    gemm(stream, oin, p_lins_W, p_lins_b, PA, RR, HH, HH, HH, HH, HH, GF_BIAS, 1.0f);
    mul_conf<<<NBLK((size_t)RR * HH), 256, 0, stream>>>(PA, CONF);
    gather_mean_bt<<<NBLK((size_t)RR * HH), 256, 0, stream>>>(PA, ROWPTR, COL, CNT, PC);
    // lin branch
    gemm(stream, oin, p_lin_W, p_lin_b, PA, RR, HH, HH, HH, HH, HH, GF_BIAS, 1.0f);
    // fin = concat([lin, t_xf, s_xf]) @ p_out_W + b  (split + accumulate)
    gemm(stream, PA, p_out_W, p_out_b, PD, RR, HH, HH, HH, HH, HH, GF_BIAS, 1.0f);
    gemm(stream, PB, p_out_W + 64 * 64, nullptr, PD, RR, HH, HH, HH, HH, HH, GF_ACCUM, 1.0f);
    gemm(stream, PC, p_out_W + 128 * 64, nullptr, PD, RR, HH, HH, HH, HH, HH, GF_ACCUM, 1.0f);
    gelu_mask_select<<<NBLK((size_t)RR * HH), 256, 0, stream>>>(PD, XF, mask, oout);
    float* tmp = oin; oin = oout; oout = tmp;
  }

  // 25. final: out @ stage2_W + b   [R,64]@[64,1]
  gemm_naive<<<NBLK(RR), 256, 0, stream>>>(oin, stage2_W, stage2_b, (float*)d_out, RR, 1, HH, GF_BIAS);
}